// SAGE_Net_84756884619999
// MI455X (gfx1250) — compile-verified
//
#include <hip/hip_runtime.h>

// ---------------- problem constants ----------------
#define N_NODES 100000
#define N_EDGES 1000000
#define D_IN    64
#define D_H1    128
#define D_H2    64

typedef __attribute__((ext_vector_type(2))) float v2f;
typedef __attribute__((ext_vector_type(8))) float v8f;

// ---------------------------------------------------------------------------
// zero a float region (graph-replay-safe workspace init)
// ---------------------------------------------------------------------------
__global__ void zero_f32(float* __restrict__ p, long long n4) {
    long long i = (long long)blockIdx.x * blockDim.x + threadIdx.x;
    if (i < n4) {
        float4 z = {0.f, 0.f, 0.f, 0.f};
        ((float4*)p)[i] = z;
    }
}

// ---------------------------------------------------------------------------
// scatter-add neighbor features: agg[dst] += x[src]; cnt[dst] += 1
// one thread per (edge, 4-float chunk); float4 gather + 4 scalar f32 atomics
// ---------------------------------------------------------------------------
__global__ void scatter_accum(const float* __restrict__ x,
                              const int* __restrict__ src,
                              const int* __restrict__ dst,
                              float* __restrict__ agg,
                              float* __restrict__ cnt,
                              int nEdges, int D, int addCount) {
    int parts = D >> 2;
    long long idx = (long long)blockIdx.x * blockDim.x + threadIdx.x;
    int e = (int)(idx / parts);
    int p = (int)(idx % parts);
    if (e >= nEdges) return;
    int s = src[e], d = dst[e];
    float4 v = ((const float4*)(x + (long long)s * D))[p];
    float* o = agg + (long long)d * D + p * 4;
    atomicAdd(o + 0, v.x);
    atomicAdd(o + 1, v.y);
    atomicAdd(o + 2, v.z);
    atomicAdd(o + 3, v.w);
    if (addCount && p == 0) atomicAdd(cnt + d, 1.0f);
}

// ---------------------------------------------------------------------------
// agg[row] *= 1 / max(cnt[row], 1)
// ---------------------------------------------------------------------------
__global__ void normalize_mean(float* __restrict__ agg,
                               const float* __restrict__ cnt,
                               int nNodes, int D) {
    int parts = D >> 2;
    long long idx = (long long)blockIdx.x * blockDim.x + threadIdx.x;
    int row = (int)(idx / parts);
    int p   = (int)(idx % parts);
    if (row >= nNodes) return;
    float inv = 1.0f / fmaxf(cnt[row], 1.0f);
    float4* ptr = (float4*)(agg + (long long)row * D) + p;
    float4 v = *ptr;
    v.x *= inv; v.y *= inv; v.z *= inv; v.w *= inv;
    *ptr = v;
}

// ---------------------------------------------------------------------------
// Fused SAGE linear:  out = act( A1 @ W1  +  A2 @ W2  + bias )
//   A1 = normalized aggregate [N, K], W1 = lin_l [K, Nout]
//   A2 = node features        [N, K], W2 = lin_r [K, Nout]
// One wave (32 threads) per 16x16 output tile, V_WMMA_F32_16X16X4_F32 over K.
// Grid: (N/16, Nout/16). All dims are exact multiples -> EXEC stays all-1s.
//
// Fragment layouts per CDNA5 ISA 7.12.2 (wave32):
//   A (16x4 f32): lane L -> row M = L&15, K = 2*(L>>4) + {0,1}   (float2 load)
//   B (4x16 f32): VGPR0 = rows {K0 | K2}, VGPR1 = rows {K1 | K3}, N = L&15
//   C/D (16x16):  VGPR i -> row m0+i (lanes 0-15) / m0+8+i (lanes 16-31)
// ---------------------------------------------------------------------------
__global__ void sage_gemm_wmma(const float* __restrict__ A1,
                               const float* __restrict__ W1,
                               const float* __restrict__ A2,
                               const float* __restrict__ W2,
                               const float* __restrict__ bias,
                               float* __restrict__ out,
                               int K, int Nout, int doRelu) {
    const int lane = threadIdx.x;          // 0..31
    const int m0   = blockIdx.x * 16;
    const int n0   = blockIdx.y * 16;
    const int mrow = m0 + (lane & 15);     // A-fragment row
    const int kgrp = (lane >> 4) * 2;      // 0 or 2
    const int ncol = n0 + (lane & 15);     // B/C/D column

    v8f acc = {};

    const float* a1row = A1 + (long long)mrow * K;
    for (int k = 0; k < K; k += 4) {
        float2 af = *(const float2*)(a1row + k + kgrp);
        v2f a; a.x = af.x; a.y = af.y;
        v2f b;
        b.x = W1[(long long)(k + kgrp + 0) * Nout + ncol];
        b.y = W1[(long long)(k + kgrp + 1) * Nout + ncol];
        acc = __builtin_amdgcn_wmma_f32_16x16x4_f32(
            /*neg_a=*/false, a, /*neg_b=*/false, b,
            /*c_mod=*/(short)0, acc, /*reuse_a=*/false, /*reuse_b=*/false);
    }

    const float* a2row = A2 + (long long)mrow * K;
    for (int k = 0; k < K; k += 4) {
        float2 af = *(const float2*)(a2row + k + kgrp);
        v2f a; a.x = af.x; a.y = af.y;
        v2f b;
        b.x = W2[(long long)(k + kgrp + 0) * Nout + ncol];
        b.y = W2[(long long)(k + kgrp + 1) * Nout + ncol];
        acc = __builtin_amdgcn_wmma_f32_16x16x4_f32(
            false, a, false, b, (short)0, acc, false, false);
    }

    const float bv = bias[ncol];
    const int mbase = m0 + ((lane >> 4) ? 8 : 0);
#pragma unroll
    for (int i = 0; i < 8; ++i) {
        float v = acc[i] + bv;
        if (doRelu) v = fmaxf(v, 0.0f);
        out[(long long)(mbase + i) * Nout + ncol] = v;
    }
}

// ---------------------------------------------------------------------------
// decode: logits[e] = dot(z[a], z[b]) over 64 dims; 8 lanes per edge
// ---------------------------------------------------------------------------
__global__ void decode_dot(const float* __restrict__ z,
                           const int* __restrict__ ei,
                           const int* __restrict__ nei,
                           float* __restrict__ out, int nE) {
    long long idx = (long long)blockIdx.x * blockDim.x + threadIdx.x;
    int sub = (int)(idx & 7);
    long long e = idx >> 3;
    if (e >= 2LL * nE) return;
    int a, b;
    if (e < nE) { a = ei[e];        b = ei[nE + e]; }
    else        { long long e2 = e - nE; a = nei[e2]; b = nei[nE + e2]; }

    const float4* za = (const float4*)(z + (long long)a * D_H2);
    const float4* zb = (const float4*)(z + (long long)b * D_H2);
    float p = 0.0f;
#pragma unroll
    for (int t = 0; t < 2; ++t) {
        float4 va = za[sub * 2 + t];
        float4 vb = zb[sub * 2 + t];
        p += va.x * vb.x + va.y * vb.y + va.z * vb.z + va.w * vb.w;
    }
    p += __shfl_xor(p, 1, 8);
    p += __shfl_xor(p, 2, 8);
    p += __shfl_xor(p, 4, 8);
    if (sub == 0) out[e] = p;
}

// ---------------------------------------------------------------------------
extern "C" void kernel_launch(void* const* d_in, const int* in_sizes, int n_in,
                              void* d_out, int out_size, void* d_ws, size_t ws_size,
                              hipStream_t stream) {
    (void)in_sizes; (void)n_in; (void)out_size; (void)ws_size;

    // inputs (setup_inputs dict order)
    const float* x    = (const float*)d_in[0];
    const int*   ei   = (const int*)  d_in[1];   // [2, E] flat
    const int*   nei  = (const int*)  d_in[2];   // [2, E] flat
    const float* W1l  = (const float*)d_in[3];
    const float* b1   = (const float*)d_in[4];
    const float* W1r  = (const float*)d_in[5];
    const float* W2l  = (const float*)d_in[6];
    const float* b2   = (const float*)d_in[7];
    const float* W2r  = (const float*)d_in[8];
    float* logits = (float*)d_out;

    // workspace layout (floats)
    float* ws   = (float*)d_ws;
    float* agg1 = ws;                                     // [N, 64]  (reused as z)
    float* cnt  = agg1 + (long long)N_NODES * D_IN;       // [N]
    float* h    = cnt  + N_NODES;                         // [N, 128]
    float* agg2 = h    + (long long)N_NODES * D_H1;       // [N, 128]
    float* z    = agg1;                                   // reuse after gemm1

    const int BLK = 256;
    // zero agg1 + cnt + h + agg2 (contiguous region)
    long long totalZero = (long long)N_NODES * (D_IN + 1 + D_H1 + D_H1); // 32.1M floats
    long long n4 = totalZero / 4;  // N_NODES*321/4 is exact (100000%4==0)
    zero_f32<<<(unsigned)((n4 + BLK - 1) / BLK), BLK, 0, stream>>>(ws, n4);

    const int* src = ei;            // edge_index[0]
    const int* dst = ei + N_EDGES;  // edge_index[1]

    // ---- layer 1 ----
    {
        long long nt = (long long)N_EDGES * (D_IN / 4);   // 16M
        scatter_accum<<<(unsigned)((nt + BLK - 1) / BLK), BLK, 0, stream>>>(
            x, src, dst, agg1, cnt, N_EDGES, D_IN, /*addCount=*/1);
        long long nn = (long long)N_NODES * (D_IN / 4);
        normalize_mean<<<(unsigned)((nn + BLK - 1) / BLK), BLK, 0, stream>>>(
            agg1, cnt, N_NODES, D_IN);
        dim3 grid(N_NODES / 16, D_H1 / 16);               // 6250 x 8
        sage_gemm_wmma<<<grid, 32, 0, stream>>>(agg1, W1l, x, W1r, b1, h,
                                                D_IN, D_H1, /*relu=*/1);
    }

    // ---- layer 2 (counts identical: same edge list) ----
    {
        long long nt = (long long)N_EDGES * (D_H1 / 4);   // 32M
        scatter_accum<<<(unsigned)((nt + BLK - 1) / BLK), BLK, 0, stream>>>(
            h, src, dst, agg2, cnt, N_EDGES, D_H1, /*addCount=*/0);
        long long nn = (long long)N_NODES * (D_H1 / 4);
        normalize_mean<<<(unsigned)((nn + BLK - 1) / BLK), BLK, 0, stream>>>(
            agg2, cnt, N_NODES, D_H1);
        dim3 grid(N_NODES / 16, D_H2 / 16);               // 6250 x 4
        sage_gemm_wmma<<<grid, 32, 0, stream>>>(agg2, W2l, h, W2r, b2, z,
                                                D_H1, D_H2, /*relu=*/0);
    }

    // ---- decode: 2E logits, 8 lanes per edge ----
    {
        long long nt = 2LL * N_EDGES * 8;                 // 16M
        decode_dot<<<(unsigned)((nt + BLK - 1) / BLK), BLK, 0, stream>>>(
            z, ei, nei, logits, N_EDGES);
    }
}